// GND_61873298866219
// MI455X (gfx1250) — compile-verified
//
#include <hip/hip_runtime.h>
#include <hip/hip_bf16.h>

// -------- problem constants (also derived from in_sizes at launch) --------
#define NHEADS 4
#define NFEAT  16
#define HF     64          // NHEADS*NFEAT floats per node
#define ROWPAD 66          // padded LDS row stride (floats) to spread banks

typedef float v2f __attribute__((ext_vector_type(2)));
typedef float v8f __attribute__((ext_vector_type(8)));

// order-preserving float <-> uint map for atomic min
__device__ __forceinline__ unsigned encf(float x) {
    unsigned b = __float_as_uint(x);
    return (b & 0x80000000u) ? ~b : (b | 0x80000000u);
}
__device__ __forceinline__ float decf(unsigned u) {
    unsigned b = (u & 0x80000000u) ? (u ^ 0x80000000u) : ~u;
    return __uint_as_float(b);
}

// -------- kernel 0: init denom + stats (ws is poisoned before timing) ------
__global__ void GND_initk(float* __restrict__ denom, int nNodes,
                          double* __restrict__ gsum, unsigned* __restrict__ gmin) {
    int i = blockIdx.x * blockDim.x + threadIdx.x;
    if (i < nNodes * NHEADS) denom[i] = 0.0f;
    if (i < NHEADS) { gsum[i] = 0.0; gmin[i] = 0xFFFFFFFFu; }
}

// -------- kernel 1: gather + f_src write + WMMA distances + stats ----------
// One wave handles a tile of 16 edges. A = (delta^2)[16 x 64], B = blockdiag
// weights [64 x 16] (col h = we^2*dw for head h), D[16x16] cols 0..3 = d[e,h].
__global__ __launch_bounds__(256) void GND_pass1(
    const float* __restrict__ nf, const int* __restrict__ eidx,
    const float* __restrict__ ew, const float* __restrict__ dwt,
    float* __restrict__ fsrc_out, float* __restrict__ wsD,
    double* __restrict__ gsum, unsigned* __restrict__ gmin,
    int E, int nTiles)
{
    __shared__ float lds[8 * 16 * ROWPAD];   // 8 waves * 16 rows * 66 floats
    __shared__ double   bsum[NHEADS];
    __shared__ unsigned bminv[NHEADS];

    const int tid       = threadIdx.x;
    const int lane      = tid & 31;
    const int waveInBlk = tid >> 5;
    const int m         = lane & 15;   // A: row M  /  B,D: column N
    const int half      = lane >> 4;   // selects K sub-pair per ISA layout

    if (tid < NHEADS) { bsum[tid] = 0.0; bminv[tid] = 0xFFFFFFFFu; }
    __syncthreads();

    // Precompute B operands for the 16 K-chunks (ISA 4x16 f32 B layout:
    // VGPR0 = rows K=0(low)/K=2(high), VGPR1 = rows K=1/K=3). Chunk t covers
    // global k = 4t..4t+3, all belonging to head t>>2; column n==head gets
    // c[k] = we[k]^2*dw[k], all other columns 0.
    v2f Bv[16];
#pragma unroll
    for (int t = 0; t < 16; ++t) {
        float b0 = 0.0f, b1 = 0.0f;
        if (m == (t >> 2)) {
            int k0 = 4 * t + 2 * half;
            float e0 = ew[k0], e1 = ew[k0 + 1];
            b0 = e0 * e0 * dwt[k0];
            b1 = e1 * e1 * dwt[k0 + 1];
        }
        Bv[t][0] = b0; Bv[t][1] = b1;
    }

    const int wavesTotal = (gridDim.x * blockDim.x) >> 5;
    const int waveId     = (blockIdx.x * blockDim.x + tid) >> 5;
    const int iters      = (nTiles + wavesTotal - 1) / wavesTotal;

    float* myLds = lds + waveInBlk * (16 * ROWPAD);
    float lsum = 0.0f;
    float lmin = 3.402823466e38f;

    for (int it = 0; it < iters; ++it) {
        int tile  = waveId + it * wavesTotal;
        bool valid = (tile < nTiles);
        int eBase = tile * 16;

        if (valid) {
            int e   = eBase + m;               // both halves load same edge
            int trg = eidx[e];
            int src = eidx[E + e];
            const float4* sp = (const float4*)(nf + (size_t)src * HF + 32 * half);
            const float4* tp = (const float4*)(nf + (size_t)trg * HF + 32 * half);
            float* row   = myLds + m * ROWPAD + 32 * half;
            float4* fout = (float4*)(fsrc_out + (size_t)e * HF + 32 * half);
#pragma unroll
            for (int j = 0; j < 8; ++j) {
                float4 s  = sp[j];
                float4 t4 = tp[j];
                float dx = t4.x - s.x, dy = t4.y - s.y;
                float dz = t4.z - s.z, dw4 = t4.w - s.w;
                row[4 * j + 0] = dx * dx;
                row[4 * j + 1] = dy * dy;
                row[4 * j + 2] = dz * dz;
                row[4 * j + 3] = dw4 * dw4;
                fout[j] = s;                   // f_src output (bulk of HBM traffic)
            }
        }
        __syncthreads();                       // publish delta^2 tiles

        if (valid) {
            v8f acc = {0.f, 0.f, 0.f, 0.f, 0.f, 0.f, 0.f, 0.f};
            // A layout 16x4 f32: lane m holds K = 2*half + {0,1} of each chunk
            const float2* rowF2 = (const float2*)(myLds + m * ROWPAD);
#pragma unroll
            for (int t = 0; t < 16; ++t) {
                float2 a2 = rowF2[2 * t + half];
                v2f A; A[0] = a2.x; A[1] = a2.y;
                acc = __builtin_amdgcn_wmma_f32_16x16x4_f32(
                        false, A, false, Bv[t], (short)0, acc, false, false);
            }
            // D layout: VGPR r, lane L -> M = r + 8*(L>>4), N = L&15
            if (m < NHEADS) {
#pragma unroll
                for (int r = 0; r < 8; ++r) {
                    int em  = r + 8 * half;
                    float v = acc[r];
                    wsD[(size_t)(eBase + em) * NHEADS + m] = v;
                    lsum += v;
                    lmin  = fminf(lmin, v);
                }
            }
        }
        __syncthreads();                       // protect LDS reuse next iter
    }

    // block-level reduction of per-head sum (double) and min
    if (m < NHEADS) {
        atomicAdd(&bsum[m], (double)lsum);
        atomicMin(&bminv[m], encf(lmin));
    }
    __syncthreads();
    if (tid < NHEADS) {
        atomicAdd(&gsum[tid], bsum[tid]);
        atomicMin(&gmin[tid], bminv[tid]);
    }
}

// -------- scalar tail (only if E % 16 != 0; E=1.6M -> unused) --------------
__global__ void GND_pass1_tail(const float* __restrict__ nf, const int* __restrict__ eidx,
                               const float* __restrict__ ew, const float* __restrict__ dwt,
                               float* __restrict__ fsrc_out, float* __restrict__ wsD,
                               double* __restrict__ gsum, unsigned* __restrict__ gmin,
                               int E, int eStart) {
    int e = eStart + blockIdx.x * blockDim.x + threadIdx.x;
    if (e >= E) return;
    int trg = eidx[e], src = eidx[E + e];
    for (int h = 0; h < NHEADS; ++h) {
        float acc = 0.0f;
        for (int f = 0; f < NFEAT; ++f) {
            int k = h * NFEAT + f;
            float sv = nf[(size_t)src * HF + k];
            float tv = nf[(size_t)trg * HF + k];
            float d  = tv - sv;
            float c  = ew[k] * ew[k] * dwt[k];
            acc += d * d * c;
            fsrc_out[(size_t)e * HF + k] = sv;
        }
        wsD[(size_t)e * NHEADS + h] = acc;
        atomicAdd(&gsum[h], (double)acc);
        atomicMin(&gmin[h], encf(acc));
    }
}

// -------- kernel 2: mean + global score max (leaky_relu is monotone) -------
__global__ void GND_pass2(const double* __restrict__ gsum, const unsigned* __restrict__ gmin,
                          float* __restrict__ meanv, float* __restrict__ smax, int E) {
    if (blockIdx.x == 0 && threadIdx.x == 0) {
        float gx = 3.402823466e38f;
        for (int h = 0; h < NHEADS; ++h) {
            float mh = (float)(gsum[h] / (double)E);
            meanv[h] = mh;
            gx = fminf(gx, decf(gmin[h]) + mh);
        }
        float lr = (gx >= 0.0f) ? gx : 0.2f * gx;
        *smax = -lr;   // == scores.max()
    }
}

// -------- kernel 3: exp(score) in place + segment-sum via L2 atomics -------
__global__ void GND_pass3(float* __restrict__ wsD, const int* __restrict__ eidx,
                          const float* __restrict__ meanv, const float* __restrict__ smax,
                          float* __restrict__ denom, int E) {
    long long i = (long long)blockIdx.x * blockDim.x + threadIdx.x;
    long long total = (long long)E * NHEADS;
    if (i >= total) return;
    int h = (int)(i & 3);
    long long e = i >> 2;
    float x  = wsD[i] + meanv[h];
    float lr = (x >= 0.0f) ? x : 0.2f * x;
    float ex = __expf(-lr - *smax);
    wsD[i] = ex;
    int t = eidx[e];
    atomicAdd(&denom[(size_t)t * NHEADS + h], ex);   // denom: 1.6MB, L2-resident
}

// -------- kernel 4: normalize --------------------------------------------
__global__ void GND_pass4(const float* __restrict__ wsD, const int* __restrict__ eidx,
                          const float* __restrict__ denom, float* __restrict__ out, int E) {
    long long i = (long long)blockIdx.x * blockDim.x + threadIdx.x;
    long long total = (long long)E * NHEADS;
    if (i >= total) return;
    int h = (int)(i & 3);
    long long e = i >> 2;
    int t = eidx[e];
    out[i] = wsD[i] / (denom[(size_t)t * NHEADS + h] + 1e-16f);
}

extern "C" void kernel_launch(void* const* d_in, const int* in_sizes, int n_in,
                              void* d_out, int out_size, void* d_ws, size_t ws_size,
                              hipStream_t stream) {
    const float* nf  = (const float*)d_in[0];   // [N, 4, 16]
    const int*  eidx = (const int*)d_in[1];     // [2, E]  (row0 = trg, row1 = src)
    const float* ew  = (const float*)d_in[2];   // [1, 4, 16]
    const float* dwt = (const float*)d_in[3];   // [1, 4, 16]

    const int E      = in_sizes[1] / 2;
    const int nNodes = in_sizes[0] / HF;

    float* att  = (float*)d_out;                    // [E,4,1]
    float* fsrc = att + (size_t)E * NHEADS;         // [E,4,16]

    // workspace layout: stats(128B) | denom[N*4] f32 | d/exp[E*4] f32 (~27.3MB)
    char*     ws    = (char*)d_ws;
    double*   gsum  = (double*)(ws + 0);
    unsigned* gmin  = (unsigned*)(ws + 32);
    float*    meanv = (float*)(ws + 48);
    float*    smax  = (float*)(ws + 64);
    float*    denom = (float*)(ws + 128);
    size_t dOff = (128 + (size_t)nNodes * NHEADS * 4 + 127) & ~(size_t)127;
    float* wsD = (float*)(ws + dOff);

    const int nTiles = E / 16;
    const int tail   = E - nTiles * 16;

    GND_initk<<<(nNodes * NHEADS + 255) / 256, 256, 0, stream>>>(denom, nNodes, gsum, gmin);

    GND_pass1<<<1250, 256, 0, stream>>>(nf, eidx, ew, dwt, fsrc, wsD, gsum, gmin, E, nTiles);
    if (tail > 0) {
        GND_pass1_tail<<<1, 32, 0, stream>>>(nf, eidx, ew, dwt, fsrc, wsD, gsum, gmin,
                                             E, nTiles * 16);
    }

    GND_pass2<<<1, 32, 0, stream>>>(gsum, gmin, meanv, smax, E);

    long long totalEH = (long long)E * NHEADS;
    int blocksEH = (int)((totalEH + 255) / 256);
    GND_pass3<<<blocksEH, 256, 0, stream>>>(wsD, eidx, meanv, smax, denom, E);
    GND_pass4<<<blocksEH, 256, 0, stream>>>(wsD, eidx, denom, att, E);
}